// SS2D_58342835749251
// MI455X (gfx1250) — compile-verified
//
#include <hip/hip_runtime.h>
#include <math.h>

typedef __attribute__((ext_vector_type(2))) float v2f;
typedef __attribute__((ext_vector_type(8))) float v8f;

#define DIMC    96
#define D_STATE 16
#define D_INNER 192
#define DT_RANK 6
#define KDIR    4
#define BATCH   2
#define HH      56
#define WWD     56
#define LL      (HH * WWD)       // 3136
#define BL      (BATCH * LL)     // 6272
#define NPROJ   (DT_RANK + 2 * D_STATE)  // 38
#define XZW     (2 * D_INNER)    // 384

// scan-index -> spatial-linear-index for direction k (H==W==56)
__device__ __forceinline__ int scan_to_spatial(int k, int l) {
    int q = l / WWD, r = l % WWD;
    if (k == 0) return l;
    if (k == 1) return q * WWD + (WWD - 1 - r);
    if (k == 2) return r * WWD + q;
    return (WWD - 1 - r) * WWD + q;   // k == 3
}

// spatial-linear-index -> scan-index for direction k
__device__ __forceinline__ int spatial_to_scan(int k, int l) {
    int q = l / WWD, r = l % WWD;     // q = h, r = w
    if (k == 0) return l;
    if (k == 1) return q * WWD + (WWD - 1 - r);   // involution
    if (k == 2) return r * WWD + q;               // involution
    return r * WWD + (WWD - 1 - q);   // k == 3 (inverse of forward map)
}

__device__ __forceinline__ float silu(float v) {
    return v / (1.f + __expf(-v));
}

// ---------------- GEMM 1: xz[BL,384] = xf[BL,96] @ W_in[96,384] -------------
// xf[row, c] = x[b, c, l]  (x is (B, 96, H, W))
__global__ void k_in_gemm(const float* __restrict__ x,
                          const float* __restrict__ Win,
                          float* __restrict__ xz) {
    int lane  = threadIdx.x & 31;
    int wave  = threadIdx.x >> 5;
    int mTile = blockIdx.x * 4 + wave;   // 392 tiles (grid.x = 98)
    int nTile = blockIdx.y;              // 24 tiles
    int m0 = mTile * 16, n0 = nTile * 16;
    int mrow = m0 + (lane & 15);
    int b = mrow / LL, l = mrow % LL;
    int nn = n0 + (lane & 15);
    int kh = (lane >> 4) * 2;
    v8f acc = {};
    for (int kb = 0; kb < DIMC; kb += 4) {
        int c0 = kb + kh;
        v2f a, bf;
        a[0]  = x[(size_t)(b * DIMC + c0    ) * LL + l];
        a[1]  = x[(size_t)(b * DIMC + c0 + 1) * LL + l];
        bf[0] = Win[(size_t)(c0    ) * XZW + nn];
        bf[1] = Win[(size_t)(c0 + 1) * XZW + nn];
        acc = __builtin_amdgcn_wmma_f32_16x16x4_f32(false, a, false, bf,
                                                    (short)0, acc, false, false);
    }
    int mb = m0 + (lane >> 4) * 8;
#pragma unroll
    for (int v = 0; v < 8; ++v)
        xz[(size_t)(mb + v) * XZW + nn] = acc[v];
}

// ---------------- depthwise 3x3 conv + bias + SiLU --------------------------
// in : xp = xz[:, 0:192]  viewed as (B, L, 192)
// out: xi (B, L, 192) d-contiguous
__global__ void k_conv(const float* __restrict__ xz,
                       const float* __restrict__ Wc,
                       const float* __restrict__ bc,
                       float* __restrict__ xi) {
    int tid = blockIdx.x * blockDim.x + threadIdx.x;
    if (tid >= BL * D_INNER) return;
    int d  = tid % D_INNER;
    int bl = tid / D_INNER;
    int b = bl / LL, l = bl % LL;
    int h = l / WWD, w = l % WWD;
    float acc = bc[d];
#pragma unroll
    for (int dy = 0; dy < 3; ++dy) {
        int hy = h + dy - 1;
        if (hy < 0 || hy >= HH) continue;
#pragma unroll
        for (int dx = 0; dx < 3; ++dx) {
            int wx = w + dx - 1;
            if (wx < 0 || wx >= WWD) continue;
            acc += Wc[d * 9 + dy * 3 + dx] *
                   xz[(size_t)(b * LL + hy * WWD + wx) * XZW + d];
        }
    }
    xi[(size_t)bl * D_INNER + d] = silu(acc);
}

// ---------------- GEMM 2: xdbl[k] = xs[k][BL,192] @ W_xproj[k][192,38] ------
// xs[k][row, d] = xi[b, scan_to_spatial(k, l), d];  N padded 38 -> 48
__global__ void k_xproj(const float* __restrict__ xi,
                        const float* __restrict__ Wxp,
                        float* __restrict__ xdbl) {
    int lane = threadIdx.x & 31;
    int wave = threadIdx.x >> 5;
    int kdir = blockIdx.z;
    int mTile = blockIdx.x * 4 + wave;   // 392
    int nTile = blockIdx.y;              // 3 (covers 48 cols)
    int m0 = mTile * 16, n0 = nTile * 16;
    int mrow = m0 + (lane & 15);
    int b = mrow / LL, l = mrow % LL;
    int lsp = scan_to_spatial(kdir, l);
    const float* arow = xi + (size_t)(b * LL + lsp) * D_INNER;
    const float* wk = Wxp + (size_t)kdir * D_INNER * NPROJ;
    int nn = n0 + (lane & 15);
    bool nok = (nn < NPROJ);
    int kh = (lane >> 4) * 2;
    v8f acc = {};
    for (int kb = 0; kb < D_INNER; kb += 4) {
        v2f a, bf;
        a[0]  = arow[kb + kh];
        a[1]  = arow[kb + kh + 1];
        bf[0] = nok ? wk[(size_t)(kb + kh    ) * NPROJ + nn] : 0.f;
        bf[1] = nok ? wk[(size_t)(kb + kh + 1) * NPROJ + nn] : 0.f;
        acc = __builtin_amdgcn_wmma_f32_16x16x4_f32(false, a, false, bf,
                                                    (short)0, acc, false, false);
    }
    if (nok) {
        int mb = m0 + (lane >> 4) * 8;
#pragma unroll
        for (int v = 0; v < 8; ++v)
            xdbl[(size_t)(kdir * BL + mb + v) * NPROJ + nn] = acc[v];
    }
}

// ---------------- GEMM 3: delta = softplus(dr @ W_dt + b_dt) ----------------
// dr[k, row, r] = xdbl[k, row, r], r < 6 ; K padded 6 -> 8
__global__ void k_delta(const float* __restrict__ xdbl,
                        const float* __restrict__ Wdt,
                        const float* __restrict__ bdt,
                        float* __restrict__ delta) {
    int lane = threadIdx.x & 31;
    int wave = threadIdx.x >> 5;
    int kdir = blockIdx.z;
    int mTile = blockIdx.x * 4 + wave;   // 392
    int nTile = blockIdx.y;              // 12
    int m0 = mTile * 16, n0 = nTile * 16;
    int mrow = m0 + (lane & 15);
    int nn = n0 + (lane & 15);
    int kh = (lane >> 4) * 2;
    v8f acc = {};
#pragma unroll
    for (int kb = 0; kb < 8; kb += 4) {
        v2f a, bf;
#pragma unroll
        for (int v = 0; v < 2; ++v) {
            int kk = kb + kh + v;
            bool kok = (kk < DT_RANK);
            a[v]  = kok ? xdbl[(size_t)(kdir * BL + mrow) * NPROJ + kk] : 0.f;
            bf[v] = kok ? Wdt[(size_t)(kdir * DT_RANK + kk) * D_INNER + nn] : 0.f;
        }
        acc = __builtin_amdgcn_wmma_f32_16x16x4_f32(false, a, false, bf,
                                                    (short)0, acc, false, false);
    }
    float bias = bdt[kdir * D_INNER + nn];
    int mb = m0 + (lane >> 4) * 8;
#pragma unroll
    for (int v = 0; v < 8; ++v) {
        float t = acc[v] + bias;
        float sp = (t > 20.f) ? t : log1pf(__expf(t));
        delta[(size_t)(kdir * BL + mb + v) * D_INNER + nn] = sp;
    }
}

// ---------------- selective scan -------------------------------------------
// 16 lanes per (k, b, d): lane-within-group = state index n. 24576 threads.
// ys[k,b,l,d] = sum_n h[n]*C[n]  +  x * Ds[k,d]
__global__ void k_scan(const float* __restrict__ xi,
                       const float* __restrict__ xdbl,
                       const float* __restrict__ delta,
                       const float* __restrict__ A_logs,
                       const float* __restrict__ Ds,
                       float* __restrict__ ys) {
    int tid = blockIdx.x * blockDim.x + threadIdx.x;
    if (tid >= KDIR * BATCH * D_INNER * D_STATE) return;
    int n     = tid & 15;
    int group = tid >> 4;                 // (k*2 + b)*192 + d
    int d     = group % D_INNER;
    int kb    = group / D_INNER;
    int b     = kb % BATCH;
    int kdir  = kb / BATCH;

    float A = -__expf(A_logs[(size_t)(kdir * D_INNER + d) * D_STATE + n]);
    float Dskip = Ds[kdir * D_INNER + d];
    float h = 0.f;

    const float* xib = xi    + (size_t)b * LL * D_INNER + d;
    const float* dlb = delta + (size_t)(kdir * BATCH + b) * LL * D_INNER + d;
    const float* xdb = xdbl  + (size_t)(kdir * BATCH + b) * LL * NPROJ;
    float*       ysb = ys    + (size_t)(kdir * BATCH + b) * LL * D_INNER + d;

    for (int l = 0; l < LL; ++l) {
        int lsp  = scan_to_spatial(kdir, l);
        float x  = xib[(size_t)lsp * D_INNER];        // broadcast across group
        float dl = dlb[(size_t)l * D_INNER];          // broadcast across group
        const float* bc = xdb + (size_t)l * NPROJ;
        __builtin_prefetch(bc + NPROJ, 0, 3);
        float Bn = bc[DT_RANK + n];                   // coalesced
        float Cn = bc[DT_RANK + D_STATE + n];         // coalesced
        float dA = __expf(dl * A);
        h = dA * h + (dl * x) * Bn;
        float p = h * Cn;
        // reduce over the 16 lanes of this group (wave32 shuffle tree)
#pragma unroll
        for (int off = 8; off > 0; off >>= 1)
            p += __shfl_xor(p, off, 16);
        if (n == 0)
            ysb[(size_t)l * D_INNER] = p + x * Dskip;
    }
}

// ---------------- combine 4 directions + SiLU gate --------------------------
__global__ void k_combine(const float* __restrict__ ys,
                          const float* __restrict__ xz,
                          float* __restrict__ y) {
    int tid = blockIdx.x * blockDim.x + threadIdx.x;
    if (tid >= BL * D_INNER) return;
    int d  = tid % D_INNER;
    int bl = tid / D_INNER;
    int b = bl / LL, l = bl % LL;
    float acc = 0.f;
#pragma unroll
    for (int k = 0; k < KDIR; ++k) {
        int ls = spatial_to_scan(k, l);
        acc += ys[(size_t)((k * BATCH + b) * LL + ls) * D_INNER + d];
    }
    float z = xz[(size_t)bl * XZW + D_INNER + d];
    y[tid] = acc * silu(z);
}

// ---------------- GEMM 4: out = y[BL,192] @ W_out[192,96], scatter NCHW -----
__global__ void k_out_gemm(const float* __restrict__ y,
                           const float* __restrict__ Wout,
                           float* __restrict__ out) {
    int lane = threadIdx.x & 31;
    int wave = threadIdx.x >> 5;
    int mTile = blockIdx.x * 4 + wave;   // 392
    int nTile = blockIdx.y;              // 6
    int m0 = mTile * 16, n0 = nTile * 16;
    int mrow = m0 + (lane & 15);
    int nn = n0 + (lane & 15);
    int kh = (lane >> 4) * 2;
    v8f acc = {};
    for (int kb = 0; kb < D_INNER; kb += 4) {
        v2f a, bf;
        a[0]  = y[(size_t)mrow * D_INNER + kb + kh];
        a[1]  = y[(size_t)mrow * D_INNER + kb + kh + 1];
        bf[0] = Wout[(size_t)(kb + kh    ) * DIMC + nn];
        bf[1] = Wout[(size_t)(kb + kh + 1) * DIMC + nn];
        acc = __builtin_amdgcn_wmma_f32_16x16x4_f32(false, a, false, bf,
                                                    (short)0, acc, false, false);
    }
    int mb = m0 + (lane >> 4) * 8;
#pragma unroll
    for (int v = 0; v < 8; ++v) {
        int row = mb + v;
        int b = row / LL, l = row % LL;
        out[(size_t)(b * DIMC + nn) * LL + l] = acc[v];
    }
}

extern "C" void kernel_launch(void* const* d_in, const int* in_sizes, int n_in,
                              void* d_out, int out_size, void* d_ws, size_t ws_size,
                              hipStream_t stream) {
    (void)in_sizes; (void)n_in; (void)out_size; (void)ws_size;
    const float* x       = (const float*)d_in[0];
    const float* W_in    = (const float*)d_in[1];
    const float* W_conv  = (const float*)d_in[2];
    const float* b_conv  = (const float*)d_in[3];
    const float* W_xproj = (const float*)d_in[4];
    const float* W_dt    = (const float*)d_in[5];
    const float* b_dt    = (const float*)d_in[6];
    const float* A_logs  = (const float*)d_in[7];
    const float* Ds      = (const float*)d_in[8];
    const float* W_out   = (const float*)d_in[9];
    float* out = (float*)d_out;

    float* ws = (float*)d_ws;
    size_t o = 0;
    float* xz    = ws + o;  o += (size_t)BL * XZW;                 // 2.41M f
    float* xi    = ws + o;  o += (size_t)BL * D_INNER;             // 1.20M f
    float* xdbl  = ws + o;  o += (size_t)KDIR * BL * NPROJ;        // 0.95M f
    float* delta = ws + o;  o += (size_t)KDIR * BL * D_INNER;      // 4.82M f
    float* ysbuf = ws + o;  o += (size_t)KDIR * BL * D_INNER;      // 4.82M f
    float* ybuf  = ws + o;  o += (size_t)BL * D_INNER;             // 1.20M f
    // total ~= 61.6 MB of workspace

    dim3 blk(128);  // 4 wave32s; one 16x16 output tile per wave
    k_in_gemm <<<dim3(98, 24),    blk, 0, stream>>>(x, W_in, xz);
    int nelt = BL * D_INNER;
    k_conv    <<<(nelt + 255) / 256, 256, 0, stream>>>(xz, W_conv, b_conv, xi);
    k_xproj   <<<dim3(98, 3, 4),  blk, 0, stream>>>(xi, W_xproj, xdbl);
    k_delta   <<<dim3(98, 12, 4), blk, 0, stream>>>(xdbl, W_dt, b_dt, delta);
    int nscan = KDIR * BATCH * D_INNER * D_STATE;   // 24576
    k_scan    <<<(nscan + 255) / 256, 256, 0, stream>>>(xi, xdbl, delta,
                                                        A_logs, Ds, ysbuf);
    k_combine <<<(nelt + 255) / 256, 256, 0, stream>>>(ysbuf, xz, ybuf);
    k_out_gemm<<<dim3(98, 6),     blk, 0, stream>>>(ybuf, W_out, out);
}